// UnlikelihoodLabelSmoothingLoss_52278341926994
// MI455X (gfx1250) — compile-verified
//
#include <hip/hip_runtime.h>
#include <hip/hip_bf16.h>

#define CLASSES   30522
#define HALFC     15261        // CLASSES / 2 (float2 elements per row, exact)
#define NTOT      2048         // number of mask rows == pred rows
#define BB        64           // batch rows
#define LL        128          // sequence length
#define MASKID    103
#define CONF      0.8f         // 1 - SMOOTHING
#define EPSSM     0.2f         // SMOOTHING

typedef float v2f __attribute__((ext_vector_type(2)));
typedef float v8f __attribute__((ext_vector_type(8)));

// ---------------------------------------------------------------------------
// Kernel A: row-major nonzero(masked == MASK_ID), zero-padded to NTOT.
// ---------------------------------------------------------------------------
__global__ void k_maskscan(const int* __restrict__ masked,
                           int* __restrict__ rows, int* __restrict__ cols) {
  __shared__ int cnt[BB];
  const int r = threadIdx.x;
  // zero-fill (faithful to jnp.nonzero(size=n) fill_value=0)
  for (int i = r; i < NTOT; i += BB) { rows[i] = 0; cols[i] = 0; }
  int c0 = 0;
  for (int l = 0; l < LL; ++l) c0 += (masked[r * LL + l] == MASKID) ? 1 : 0;
  cnt[r] = c0;
  __syncthreads();
  int off = 0;
  for (int i = 0; i < r; ++i) off += cnt[i];
  for (int l = 0; l < LL; ++l) {
    if (masked[r * LL + l] == MASKID) {
      if (off < NTOT) { rows[off] = r; cols[off] = l; }
      ++off;
    }
  }
}

// ---------------------------------------------------------------------------
// Kernel B: single streaming pass per row: online max / sum(exp) / sum(pred).
// This is the bandwidth-bound kernel: 250 MB read once (~10.7 us at 23.3 TB/s).
// ---------------------------------------------------------------------------
__global__ void k_rowstats(const float* __restrict__ pred,
                           float* __restrict__ logZ,
                           float* __restrict__ sumpred) {
  const int n = blockIdx.x;
  const int tid = threadIdx.x;
  const float2* row = (const float2*)(pred + (size_t)n * CLASSES);  // 8B aligned
  float m = -INFINITY, s = 0.0f, sp = 0.0f;
  for (int i = tid; i < HALFC; i += 256) {
    float2 v = row[i];
    float x = v.x;
    sp += x;
    if (x > m) { s = s * __expf(m - x) + 1.0f; m = x; } else { s += __expf(x - m); }
    x = v.y;
    sp += x;
    if (x > m) { s = s * __expf(m - x) + 1.0f; m = x; } else { s += __expf(x - m); }
  }
  __shared__ float sm[256], ss[256], spd[256];
  sm[tid] = m; ss[tid] = s; spd[tid] = sp;
  __syncthreads();
  for (int st = 128; st > 0; st >>= 1) {
    if (tid < st) {
      const float m2 = sm[tid + st], s2 = ss[tid + st];
      const float M = fmaxf(sm[tid], m2);
      ss[tid] = ss[tid] * __expf(sm[tid] - M) + s2 * __expf(m2 - M);
      sm[tid] = M;
      spd[tid] += spd[tid + st];
    }
    __syncthreads();
  }
  if (tid == 0) {
    logZ[n] = sm[0] + __logf(ss[0]);
    sumpred[n] = spd[0];
  }
}

// ---------------------------------------------------------------------------
// Kernel C: per mask-row — sparse negative-target construction (exactly the
// reference's neg_weight / 2-gram-roll logic) + gathers + per-row MLE term.
// ---------------------------------------------------------------------------
__global__ void k_rowloss(const float* __restrict__ pred,
                          const int* __restrict__ target,
                          const int* __restrict__ tok,
                          const float* __restrict__ amask,
                          const float* __restrict__ weight,
                          const int* __restrict__ rows,
                          const int* __restrict__ cols,
                          const float* __restrict__ logZ,
                          const float* __restrict__ sumpred,
                          float* __restrict__ wl,
                          float* __restrict__ wv,
                          float* __restrict__ ul) {
  const int n = blockIdx.x;
  const int l = threadIdx.x;                 // 0..127
  const int r = rows[n];
  const int c = cols[n];

  const float am = amask[r * LL + l];
  const int   id = tok[r * LL + l];
  const float ai = am * (float)id;           // exact for id < 2^24

  __shared__ float sAI[LL];
  __shared__ float sNC[LL];
  sAI[l] = ai;
  __syncthreads();
  const float center = sAI[c];
  const float nw = am - ((ai == center) ? 1.0f : 0.0f);
  const float nc = ai * nw;
  sNC[l] = nc;
  __syncthreads();

  float w;
  if (c > 0) {
    const float prev = sNC[c - 1];
    // next_mask = roll(prev_mask, 1): position l sees prev_mask[(l-1) mod L]
    const float pm = (sNC[(l + LL - 1) & (LL - 1)] == prev) ? 1.0f : 0.0f;
    w = nw * (1.0f + pm);
  } else {
    w = nw;
  }

  const float lz = logZ[n];
  float term = 0.0f;
  if (w != 0.0f) {
    const int v = (int)ai;                   // class id (== nc2 cast, when w != 0)
    const float x = pred[(size_t)n * CLASSES + v];
    const float p = __expf(x - lz);
    const float omp = fmaxf(1.0f - p, 1e-5f);
    term = -__logf(omp) * w;
  }

  __shared__ float red[LL];
  red[l] = term;
  __syncthreads();
  for (int st = LL / 2; st > 0; st >>= 1) {
    if (l < st) red[l] += red[l + st];
    __syncthreads();
  }

  if (l == 0) {
    const int t = target[n];
    const float xt = pred[(size_t)n * CLASSES + t];
    const float logpt = xt - lz;
    const float sumlogp = sumpred[n] - (float)CLASSES * lz;
    const float wt = weight[t];
    const float mle = wt * (-(CONF * logpt)
                            - (EPSSM / (float)(CLASSES - 1)) * (sumlogp - logpt));
    wl[n] = mle;
    wv[n] = wt;
    ul[n] = red[0];
  }
}

// ---------------------------------------------------------------------------
// Kernel D: deterministic reduction of the three 2048-vectors via
// v_wmma_f32_16x16x4_f32 (A = 16x4 fp32 tiles, B = ones, fp32 accumulate).
// One wave32, EXEC all ones.
// ---------------------------------------------------------------------------
__device__ __forceinline__ float wmma_sum2048(const float* __restrict__ p,
                                              float* tmp) {
  const int lane = threadIdx.x;
  const int mrow = lane & 15;
  const int koff = (lane < 16) ? 0 : 2;      // A layout: lanes 0-15 K=0,1; 16-31 K=2,3
  v2f b; b[0] = 1.0f; b[1] = 1.0f;           // all-ones B (layout-invariant)
  v8f acc = {0.f, 0.f, 0.f, 0.f, 0.f, 0.f, 0.f, 0.f};
  for (int t = 0; t < NTOT / 64; ++t) {      // 32 accumulate steps
    const float* base = p + t * 64 + mrow * 4 + koff;
    v2f a; a[0] = base[0]; a[1] = base[1];
    acc = __builtin_amdgcn_wmma_f32_16x16x4_f32(false, a, false, b, 0, acc,
                                                false, false);
  }
  // Each D entry at (M, anyN) holds the full row-M sum. Lane 0 carries M=0..7,
  // lane 16 carries M=8..15 (both at column N=0).
  const float local = acc[0] + acc[1] + acc[2] + acc[3]
                    + acc[4] + acc[5] + acc[6] + acc[7];
  __syncthreads();
  tmp[lane] = local;
  __syncthreads();
  return tmp[0] + tmp[16];
}

__global__ void k_final(const float* __restrict__ wl,
                        const float* __restrict__ wv,
                        const float* __restrict__ ul,
                        float* __restrict__ out) {
  __shared__ float tmp[32];
  const float S1 = wmma_sum2048(wl, tmp);    // sum weighted MLE loss
  const float S2 = wmma_sum2048(wv, tmp);    // sum of weights
  const float S3 = wmma_sum2048(ul, tmp);    // sum unlikelihood terms
  if (threadIdx.x == 0) {
    const float mle = S1 / S2;               // mean(wl)*N/sum(w) == sum/sumw
    const float u = S3 / (float)NTOT;
    out[0] = mle + u;                        // RANK_ALPHA == 1
    out[1] = mle;
    out[2] = u;
  }
}

// ---------------------------------------------------------------------------
extern "C" void kernel_launch(void* const* d_in, const int* in_sizes, int n_in,
                              void* d_out, int out_size, void* d_ws, size_t ws_size,
                              hipStream_t stream) {
  (void)in_sizes; (void)n_in; (void)out_size; (void)ws_size;
  const float* pred   = (const float*)d_in[0];
  const int*   target = (const int*)d_in[1];
  const int*   tok    = (const int*)d_in[2];
  const float* amask  = (const float*)d_in[3];
  const int*   masked = (const int*)d_in[4];
  const float* weight = (const float*)d_in[5];

  char* ws = (char*)d_ws;
  int*   rows    = (int*)(ws + 0 * NTOT * 4);
  int*   cols    = (int*)(ws + 1 * NTOT * 4);
  float* logZ    = (float*)(ws + 2 * NTOT * 4);
  float* sumpred = (float*)(ws + 3 * NTOT * 4);
  float* wl      = (float*)(ws + 4 * NTOT * 4);
  float* wv      = (float*)(ws + 5 * NTOT * 4);
  float* ul      = (float*)(ws + 6 * NTOT * 4);
  float* out     = (float*)d_out;

  k_maskscan<<<dim3(1),    dim3(BB),  0, stream>>>(masked, rows, cols);
  k_rowstats<<<dim3(NTOT), dim3(256), 0, stream>>>(pred, logZ, sumpred);
  k_rowloss <<<dim3(NTOT), dim3(LL),  0, stream>>>(pred, target, tok, amask, weight,
                                                   rows, cols, logZ, sumpred,
                                                   wl, wv, ul);
  k_final   <<<dim3(1),    dim3(32),  0, stream>>>(wl, wv, ul, out);
}